// FrustumToBEV_18081812316814
// MI455X (gfx1250) — compile-verified
//
#include <hip/hip_runtime.h>
#include <math.h>

// ---------------------------------------------------------------------------
// FrustumToBEV (Lift-Splat voxel pooling) for gfx1250 / MI455X.
//
//  in[0] frustum        [2,6,48,28,60,64] f32   (C innermost -> coalesced gathers)
//  in[1] intrinsics     [2,6,3,3]         f32
//  in[2] lidar_to_sensor[2,6,4,4]         f32
//  out                  [2,64,128,128]    f32   (B,C,Y,X)
//
// Stage 1: P[b,n] (3x4) = K * inv(L2S)[0:3,:]            -> d_ws (576 B)
// Stage 2: wave = (b, y, x0..x0+3). Per camera n:
//          one V_WMMA_F32_16X16X4_F32 computes uvd for 16 samples
//          (4 x-positions x 4 z-levels), redistributed via LDS, then
//          trilinear 8-corner gather of 64 channels (2 ch/lane, b64 loads),
//          accumulated over n and z in registers. Plain stores, no atomics.
// ---------------------------------------------------------------------------

typedef float v2f __attribute__((ext_vector_type(2)));
typedef float v8f __attribute__((ext_vector_type(8)));

#define NC 6
#define DDIM 48
#define HDIM 28
#define WDIM 60
#define CDIM 64

// ---------------- Stage 1: per-view projection matrices --------------------
__global__ void pmat_kernel(const float* __restrict__ Kin,
                            const float* __restrict__ Lin,
                            float* __restrict__ Pout) {
    int t = blockIdx.x * blockDim.x + threadIdx.x;
    if (t >= 2 * NC) return;
    const float* m = Lin + t * 16;
    float inv[16];
    // Adjugate 4x4 inverse; works for row-major since inv(M^T) = inv(M)^T.
    inv[0]  =  m[5]*m[10]*m[15] - m[5]*m[11]*m[14] - m[9]*m[6]*m[15] + m[9]*m[7]*m[14] + m[13]*m[6]*m[11] - m[13]*m[7]*m[10];
    inv[4]  = -m[4]*m[10]*m[15] + m[4]*m[11]*m[14] + m[8]*m[6]*m[15] - m[8]*m[7]*m[14] - m[12]*m[6]*m[11] + m[12]*m[7]*m[10];
    inv[8]  =  m[4]*m[9]*m[15]  - m[4]*m[11]*m[13] - m[8]*m[5]*m[15] + m[8]*m[7]*m[13] + m[12]*m[5]*m[11] - m[12]*m[7]*m[9];
    inv[12] = -m[4]*m[9]*m[14]  + m[4]*m[10]*m[13] + m[8]*m[5]*m[14] - m[8]*m[6]*m[13] - m[12]*m[5]*m[10] + m[12]*m[6]*m[9];
    inv[1]  = -m[1]*m[10]*m[15] + m[1]*m[11]*m[14] + m[9]*m[2]*m[15] - m[9]*m[3]*m[14] - m[13]*m[2]*m[11] + m[13]*m[3]*m[10];
    inv[5]  =  m[0]*m[10]*m[15] - m[0]*m[11]*m[14] - m[8]*m[2]*m[15] + m[8]*m[3]*m[14] + m[12]*m[2]*m[11] - m[12]*m[3]*m[10];
    inv[9]  = -m[0]*m[9]*m[15]  + m[0]*m[11]*m[13] + m[8]*m[1]*m[15] - m[8]*m[3]*m[13] - m[12]*m[1]*m[11] + m[12]*m[3]*m[9];
    inv[13] =  m[0]*m[9]*m[14]  - m[0]*m[10]*m[13] - m[8]*m[1]*m[14] + m[8]*m[2]*m[13] + m[12]*m[1]*m[10] - m[12]*m[2]*m[9];
    inv[2]  =  m[1]*m[6]*m[15]  - m[1]*m[7]*m[14]  - m[5]*m[2]*m[15] + m[5]*m[3]*m[14] + m[13]*m[2]*m[7]  - m[13]*m[3]*m[6];
    inv[6]  = -m[0]*m[6]*m[15]  + m[0]*m[7]*m[14]  + m[4]*m[2]*m[15] - m[4]*m[3]*m[14] - m[12]*m[2]*m[7]  + m[12]*m[3]*m[6];
    inv[10] =  m[0]*m[5]*m[15]  - m[0]*m[7]*m[13]  - m[4]*m[1]*m[15] + m[4]*m[3]*m[13] + m[12]*m[1]*m[7]  - m[12]*m[3]*m[5];
    inv[14] = -m[0]*m[5]*m[14]  + m[0]*m[6]*m[13]  + m[4]*m[1]*m[14] - m[4]*m[2]*m[13] - m[12]*m[1]*m[6]  + m[12]*m[2]*m[5];
    inv[3]  = -m[1]*m[6]*m[11]  + m[1]*m[7]*m[10]  + m[5]*m[2]*m[11] - m[5]*m[3]*m[10] - m[9]*m[2]*m[7]   + m[9]*m[3]*m[6];
    inv[7]  =  m[0]*m[6]*m[11]  - m[0]*m[7]*m[10]  - m[4]*m[2]*m[11] + m[4]*m[3]*m[10] + m[8]*m[2]*m[7]   - m[8]*m[3]*m[6];
    inv[11] = -m[0]*m[5]*m[11]  + m[0]*m[7]*m[9]   + m[4]*m[1]*m[11] - m[4]*m[3]*m[9]  - m[8]*m[1]*m[7]   + m[8]*m[3]*m[5];
    inv[15] =  m[0]*m[5]*m[10]  - m[0]*m[6]*m[9]   - m[4]*m[1]*m[10] + m[4]*m[2]*m[9]  + m[8]*m[1]*m[6]   - m[8]*m[2]*m[5];
    float det = m[0]*inv[0] + m[1]*inv[4] + m[2]*inv[8] + m[3]*inv[12];
    float id  = 1.0f / det;
    const float* k = Kin + t * 9;
#pragma unroll
    for (int i = 0; i < 3; ++i)
#pragma unroll
        for (int j = 0; j < 4; ++j)
            Pout[t * 12 + i * 4 + j] =
                (k[i*3+0]*inv[0*4+j] + k[i*3+1]*inv[1*4+j] + k[i*3+2]*inv[2*4+j]) * id;
}

// ---------------- Stage 2: projection (WMMA) + trilinear gather ------------
__global__ __launch_bounds__(256) void bev_kernel(const float* __restrict__ fr,
                                                  const float* __restrict__ Pm,
                                                  float* __restrict__ out) {
    __shared__ float s_uvd[8][48];          // per-wave: 16 points x (u,v,d)

    const int tid  = threadIdx.x;
    const int lane = tid & 31;
    const int wib  = tid >> 5;              // wave in block (8 waves)
    const int wg   = blockIdx.x * 8 + wib;  // 0..8191 global wave id
    const int b    = wg >> 12;              // 4096 waves per batch (128 y * 32 xb)
    const int rem  = wg & 4095;
    const int y    = rem >> 5;              // 0..127
    const int x0   = (rem & 31) << 2;       // x block of 4

    const int nn = lane & 15;               // column / point index within half
    const int hi = lane >> 4;

    // A-matrix (16 points x 4 homog), ISA 32-bit 16x4 layout:
    //   lanes 0-15: VGPR0=K0, VGPR1=K1 ; lanes 16-31: VGPR0=K2, VGPR1=K3
    // point m: x = x0 + (m&3), z-level = m>>2
    const float gy  = -47.625f + 0.75f * (float)y;
    const float gxm = -47.625f + 0.75f * (float)(x0 + (nn & 3));
    const float gzm = -3.0f + 2.0f * (float)(nn >> 2);
    v2f amat;
    amat.x = hi ? gzm : gxm;   // K=2 : K=0
    amat.y = hi ? 1.0f : gy;   // K=3 : K=1

    float2 acc[4];
#pragma unroll
    for (int p = 0; p < 4; ++p) { acc[p].x = 0.0f; acc[p].y = 0.0f; }

    const int ch = lane * 2;                // channels {ch, ch+1}
    float* s = &s_uvd[wib][0];

#pragma unroll 1
    for (int n = 0; n < NC; ++n) {
        const float4* Pr = (const float4*)(Pm + (b * NC + n) * 12);
        float4 r0 = Pr[0], r1 = Pr[1], r2 = Pr[2];
        // B = P^T (4x16), columns N>=3 zero-padded. Layout assumed symmetric
        // to documented A layout: VGPR0 holds K0 (lanes 0-15)/K2 (16-31),
        // VGPR1 holds K1/K3.
        float4 rr;
        rr.x = (nn==0)?r0.x:(nn==1)?r1.x:(nn==2)?r2.x:0.0f;
        rr.y = (nn==0)?r0.y:(nn==1)?r1.y:(nn==2)?r2.y:0.0f;
        rr.z = (nn==0)?r0.z:(nn==1)?r1.z:(nn==2)?r2.z:0.0f;
        rr.w = (nn==0)?r0.w:(nn==1)?r1.w:(nn==2)?r2.w:0.0f;
        v2f bmat;
        bmat.x = hi ? rr.z : rr.x;   // K=2 : K=0
        bmat.y = hi ? rr.w : rr.y;   // K=3 : K=1

#if __has_builtin(__builtin_amdgcn_wmma_f32_16x16x4_f32)
        v8f cz = {0.f, 0.f, 0.f, 0.f, 0.f, 0.f, 0.f, 0.f};
        v8f dm = __builtin_amdgcn_wmma_f32_16x16x4_f32(
            /*neg_a=*/false, amat, /*neg_b=*/false, bmat,
            /*c_mod=*/(short)0, cz, /*reuse_a=*/false, /*reuse_b=*/false);
        // D layout: VGPR r, lanes 0-15 -> (M=r, N=lane), lanes 16-31 -> (M=r+8).
        // Only N=0..2 (= u,v,d) are meaningful: lanes with nn<3 scatter to LDS.
        if (nn < 3) {
#pragma unroll
            for (int r = 0; r < 8; ++r) {
                int mm = r + (hi << 3);
                s[mm * 3 + nn] = dm[r];
            }
        }
#else
        // Scalar fallback (also used by host pass): lane<16 computes point nn.
        if (lane < 16) {
            const float* pp = Pm + (b * NC + n) * 12;
            float gx2 = -47.625f + 0.75f * (float)(x0 + (nn & 3));
            float gz2 = -3.0f + 2.0f * (float)(nn >> 2);
#pragma unroll
            for (int i = 0; i < 3; ++i)
                s[nn * 3 + i] = pp[i*4+0]*gx2 + pp[i*4+1]*gy + pp[i*4+2]*gz2 + pp[i*4+3];
        }
#endif
        // In-wave LDS is in-order; wait for DS stores, fence compiler ordering.
        asm volatile("s_wait_dscnt 0" ::: "memory");

        const float* fb = fr + (size_t)(b * NC + n) * (DDIM * HDIM * WDIM * CDIM);
#pragma unroll
        for (int m = 0; m < 16; ++m) {
            const int px = m & 3;
            float u  = s[m * 3 + 0];
            float v  = s[m * 3 + 1];
            float dd = s[m * 3 + 2];
            float invd = __builtin_amdgcn_rcpf(dd);
            // grid_sample align_corners=True source coords
            float ix = u * invd * (59.0f / 60.0f);
            float iy = v * invd * (27.0f / 28.0f);
            float iz = (dd - 2.0f) * (47.0f / 48.0f);
            float fx = floorf(ix), fy = floorf(iy), fz = floorf(iz);
            float tx = ix - fx, ty = iy - fy, tz = iz - fz;
            // Float compares: NaN/Inf (d ~ 0 / behind camera) -> all invalid.
            bool vx0 = (fx >=  0.0f) & (fx <= 59.0f);
            bool vx1 = (fx >= -1.0f) & (fx <= 58.0f);
            bool vy0 = (fy >=  0.0f) & (fy <= 27.0f);
            bool vy1 = (fy >= -1.0f) & (fy <= 26.0f);
            bool vz0 = (fz >=  0.0f) & (fz <= 47.0f);
            bool vz1 = (fz >= -1.0f) & (fz <= 46.0f);
            if (!((vx0 | vx1) & (vy0 | vy1) & (vz0 | vz1))) continue;  // uniform
            int xi = (int)fx, yi = (int)fy, zi = (int)fz;
            float wx0 = 1.0f - tx, wy0 = 1.0f - ty, wz0 = 1.0f - tz;
#pragma unroll
            for (int dz = 0; dz < 2; ++dz) {
                bool vz = dz ? vz1 : vz0; float wz = dz ? tz : wz0;
#pragma unroll
                for (int dy = 0; dy < 2; ++dy) {
                    bool vy = dy ? vy1 : vy0; float wyz = (dy ? ty : wy0) * wz;
#pragma unroll
                    for (int dx = 0; dx < 2; ++dx) {
                        bool vx = dx ? vx1 : vx0;
                        if (vx & vy & vz) {                       // uniform branch
                            float w = (dx ? tx : wx0) * wyz;
                            size_t idx = ((size_t)(((zi + dz) * HDIM + (yi + dy)) * WDIM
                                                   + (xi + dx))) * CDIM + ch;
                            float2 f = *(const float2*)(fb + idx); // 256B/wave coalesced
                            acc[px].x += w * f.x;
                            acc[px].y += w * f.y;
                        }
                    }
                }
            }
        }
    }

    // out[b, c, y, x]: every element covered exactly once.
#pragma unroll
    for (int j = 0; j < 2; ++j) {
        size_t cb = (((size_t)(b * CDIM + ch + j) * 128 + y) * 128) + x0;
#pragma unroll
        for (int p = 0; p < 4; ++p)
            out[cb + p] = j ? acc[p].y : acc[p].x;
    }
}

extern "C" void kernel_launch(void* const* d_in, const int* in_sizes, int n_in,
                              void* d_out, int out_size, void* d_ws, size_t ws_size,
                              hipStream_t stream) {
    const float* frustum    = (const float*)d_in[0];
    const float* intrinsics = (const float*)d_in[1];
    const float* l2s        = (const float*)d_in[2];
    float* Pws = (float*)d_ws;                     // 12 * 12 floats = 576 B
    pmat_kernel<<<1, 32, 0, stream>>>(intrinsics, l2s, Pws);
    bev_kernel<<<1024, 256, 0, stream>>>(frustum, Pws, (float*)d_out);
}